// DFsmn_21586505629747
// MI455X (gfx1250) — compile-verified
//
#include <hip/hip_runtime.h>
#include <hip/hip_bf16.h>
#include <stdint.h>

typedef int v4i __attribute__((ext_vector_type(4)));

#define GAS __attribute__((address_space(1)))
#define LAS __attribute__((address_space(3)))

// ---- CDNA5 async global -> LDS copy (16B per lane) ------------------------
__device__ __forceinline__ void async_copy_b128(const float* g, float* l) {
#if __has_builtin(__builtin_amdgcn_global_load_async_to_lds_b128)
  __builtin_amdgcn_global_load_async_to_lds_b128((GAS v4i*)g, (LAS v4i*)l, 0, 0);
#else
  unsigned int loff = (unsigned int)(unsigned long long)(LAS float*)l;
  asm volatile("global_load_async_to_lds_b128 %0, %1, off"
               :: "v"(loff), "v"(g)
               : "memory");
#endif
}

template <int N>
__device__ __forceinline__ void wait_asynccnt() {
#if __has_builtin(__builtin_amdgcn_s_wait_asynccnt)
  __builtin_amdgcn_s_wait_asynccnt(N);
#else
  asm volatile("s_wait_asynccnt %0" :: "i"(N) : "memory");
#endif
}

// ---- problem constants -----------------------------------------------------
constexpr int D_V   = 512;
constexpr int T_LEN = 1024;
constexpr int CH    = 32;            // time rows per LDS chunk
constexpr int NCH   = T_LEN / CH;    // 32 chunks
constexpr int DB    = 128;           // d-columns per block
constexpr int NBUF  = 3;             // triple-buffered LDS pipeline

__global__ __launch_bounds__(DB) void dfsmn_kernel(
    const float* __restrict__ v, const float* __restrict__ lfil,
    const float* __restrict__ rfil, float* __restrict__ out)
{
  __shared__ float smem[NBUF * CH * DB];   // 48 KB

  const int tid = threadIdx.x;             // d within slice
  const int bh  = blockIdx.x >> 2;         // 0..63  (b*H + h)
  const int d0  = (blockIdx.x & 3) * DB;   // d-slice origin
  const int d   = d0 + tid;

  const float* vbase = v   + (size_t)bh * T_LEN * D_V + d0;
  float*       obase = out + (size_t)bh * T_LEN * D_V + d0;

  // ---- per-channel coefficients in registers ----
  const float c0 = 1.0f + lfil[d];                 // l_filter[0]
  const float a1 = lfil[1 * D_V + d];
  const float a2 = lfil[2 * D_V + d];
  const float a3 = lfil[3 * D_V + d];
  const float a4 = lfil[4 * D_V + d];
  const float a5 = lfil[5 * D_V + d];
  const float a6 = lfil[6 * D_V + d];
  const float a7 = lfil[7 * D_V + d];
  const float a8 = lfil[8 * D_V + d];
  const float a9 = lfil[9 * D_V + d];
  const float r0 = rfil[d];
  const float r1 = rfil[D_V + d];

  // ---- 9-deep output history (h1 = p[t-1] ... h9 = p[t-9]) ----
  float h1 = 0.f, h2 = 0.f, h3 = 0.f, h4 = 0.f, h5 = 0.f,
        h6 = 0.f, h7 = 0.f, h8 = 0.f, h9 = 0.f;

  // async issue layout: instruction i moves rows [i*4 .. i*4+3] of the chunk;
  // lane -> (row = i*4 + tid/32, col = (tid%32)*4) : 16B per lane, coalesced.
  const int grow = tid >> 5;
  const int gcol = (tid & 31) << 2;

  auto issue = [&](int k) {
    const float* g0 = vbase + (size_t)(k * CH) * D_V;
    float*       l0 = smem + (k % NBUF) * (CH * DB);
#pragma unroll
    for (int i = 0; i < 8; ++i) {
      const int row = i * 4 + grow;
      async_copy_b128(g0 + (size_t)row * D_V + gcol, l0 + row * DB + gcol);
    }
  };

  // prologue: two chunks in flight
  issue(0);
  issue(1);

  float nxt1 = 0.f, nxt2 = 0.f;   // window: v[t], v[t+1] on loop entry

  for (int k = 0; k < NCH; ++k) {
    if (k + 2 < NCH) { issue(k + 2); wait_asynccnt<8>(); }  // chunks <= k+1 landed
    else             { wait_asynccnt<0>(); }
    __syncthreads();

    const float* cur = smem + ( k      % NBUF) * (CH * DB);
    const float* nxt = smem + ((k + 1) % NBUF) * (CH * DB);

    if (k == 0) { nxt1 = cur[tid]; nxt2 = cur[DB + tid]; }

    const int t0 = k * CH;
#pragma unroll
    for (int tt = 0; tt < CH; ++tt) {
      const int t = t0 + tt;
      const float vt  = nxt1;     // v[t]
      const float vt1 = nxt2;     // v[t+1]
      float vt2;                  // v[t+2] (zero-padded past T)
      if (tt < CH - 2) {
        vt2 = cur[(tt + 2) * DB + tid];
      } else {
        const float cand = nxt[(tt + 2 - CH) * DB + tid];
        vt2 = (t + 2 < T_LEN) ? cand : 0.0f;
      }

      // base term
      float acc = c0 * vt;
      acc = fmaf(r0, vt1, acc);
      acc = fmaf(r1, vt2, acc);
      // oldest history first: keeps only the final FMA on the t->t+1 chain
      acc = fmaf(a9, h9, acc);
      acc = fmaf(a8, h8, acc);
      acc = fmaf(a7, h7, acc);
      acc = fmaf(a6, h6, acc);
      acc = fmaf(a5, h5, acc);
      acc = fmaf(a4, h4, acc);
      acc = fmaf(a3, h3, acc);
      acc = fmaf(a2, h2, acc);
      const float p = fmaf(a1, h1, acc);

      h9 = h8; h8 = h7; h7 = h6; h6 = h5; h5 = h4;
      h4 = h3; h3 = h2; h2 = h1; h1 = p;

      __builtin_nontemporal_store(p, obase + (size_t)t * D_V + tid);

      nxt1 = vt1;
      nxt2 = vt2;
    }
    __syncthreads();   // buffer k%3 may now be re-targeted by future issues
  }
}

extern "C" void kernel_launch(void* const* d_in, const int* in_sizes, int n_in,
                              void* d_out, int out_size, void* d_ws, size_t ws_size,
                              hipStream_t stream) {
  (void)in_sizes; (void)n_in; (void)out_size; (void)d_ws; (void)ws_size;
  const float* v  = (const float*)d_in[0];   // (16,4,1024,512) f32
  const float* lf = (const float*)d_in[1];   // (10,512) f32
  const float* rf = (const float*)d_in[2];   // (2,512)  f32
  float* out = (float*)d_out;                // (16,4,1024,512) f32

  dim3 grid(16 * 4 * (D_V / DB));            // 256 blocks
  dim3 block(DB);                            // 128 threads = 4 waves
  dfsmn_kernel<<<grid, block, 0, stream>>>(v, lf, rf, out);
}